// CausalityMapBlock_74217034874995
// MI455X (gfx1250) — compile-verified
//
#include <hip/hip_runtime.h>
#include <hip/hip_bf16.h>

#define EPS  1e-8f
#define BDIM 2
#define CDIM 128
#define FDIM 49

typedef __attribute__((ext_vector_type(2))) float v2f;
typedef __attribute__((ext_vector_type(8))) float v8f;

// ---------------- Kernel 1: per-batch global max over C*F elements ----------------
__global__ void k_batch_max(const float* __restrict__ x, float* __restrict__ maxf) {
    __shared__ float red[256];
    const int b = blockIdx.x;
    const float* xb = x + (size_t)b * CDIM * FDIM;
    float m = -1e30f;
    for (int i = threadIdx.x; i < CDIM * FDIM; i += blockDim.x)
        m = fmaxf(m, xb[i]);
    red[threadIdx.x] = m;
    __syncthreads();
    for (int s = 128; s > 0; s >>= 1) {
        if (threadIdx.x < s) red[threadIdx.x] = fmaxf(red[threadIdx.x], red[threadIdx.x + s]);
        __syncthreads();
    }
    if (threadIdx.x == 0) maxf[b] = red[0];
}

// ------------- Kernel 2: per-channel moments S1, S2 and Lehmer denominator -------------
__global__ void k_channel_sums(const float* __restrict__ x, const float* __restrict__ maxf,
                               float* __restrict__ S1, float* __restrict__ S2,
                               float* __restrict__ Ld) {
    __shared__ float r1[64];
    __shared__ float r2[64];
    const int bc = blockIdx.x;           // b*C + c
    const int b  = bc / CDIM;
    const float inv = 1.0f / (maxf[b] + EPS);
    const int t = threadIdx.x;
    float v = 0.0f;
    if (t < FDIM) v = fabsf(x[(size_t)bc * FDIM + t] * inv);
    r1[t] = v;
    r2[t] = v * v;
    __syncthreads();
    for (int s = 32; s > 0; s >>= 1) {
        if (t < s) { r1[t] += r1[t + s]; r2[t] += r2[t + s]; }
        __syncthreads();
    }
    if (t == 0) {
        const float s1 = r1[0], s2 = r2[0];
        S1[bc] = s1;
        S2[bc] = s2;
        // num_den = sum (a+eps)^2 + eps ; den_den = sum (a+eps) + eps
        const float nd = s2 + 2.0f * EPS * s1 + (float)FDIM * EPS * EPS + EPS;
        const float dd = s1 + (float)FDIM * EPS + EPS;
        Ld[bc] = nd / dd + EPS;          // lehmer_denominator[b,c]
    }
}

// ------------- Kernel 3: 16x16 output tiles via V_WMMA_F32_16X16X4_F32 -------------
// One wave (32 threads) per tile; grid = (C/16, C/16, B).
__global__ void __launch_bounds__(32)
k_causality_wmma(const float* __restrict__ S1, const float* __restrict__ S2,
                 const float* __restrict__ Ld, float* __restrict__ out) {
    const int lane = threadIdx.x;        // 0..31
    const int half = lane >> 4;          // 0: lanes 0-15, 1: lanes 16-31
    const int idx  = lane & 15;
    const int n0 = blockIdx.x * 16;
    const int m0 = blockIdx.y * 16;
    const int b  = blockIdx.z;

    const float s1m = S1[b * CDIM + m0 + idx];
    const float s2m = S2[b * CDIM + m0 + idx];
    const float s1n = S1[b * CDIM + n0 + idx];
    const float s2n = S2[b * CDIM + n0 + idx];
    const float ldn = Ld[b * CDIM + n0 + idx];

    const float CNUM = (float)(FDIM * FDIM) * EPS * EPS + EPS;  // F^2*eps^2 + eps
    const float CDEN = (float)(FDIM * FDIM) * EPS + EPS;        // F^2*eps   + eps

    // A 16x4 layout (32-bit): lanes 0-15 -> (K0,K1) for row M=idx; lanes 16-31 -> (K2,K3)
    // B 4x16 layout: VGPR0: lanes 0-15 = row K0, lanes 16-31 = row K2;
    //                VGPR1: lanes 0-15 = row K1, lanes 16-31 = row K3
    v2f aN, bN, aD, bD;
    if (half == 0) {
        aN.x = s2m;  aN.y = s1m;                 // K0=S2m, K1=S1m
        aD.x = s1m;  aD.y = 1.0f;                // K0=S1m, K1=1
        bN.x = s2n;  bN.y = 2.0f * EPS * s1n;    // K0=S2n, K1=2eps*S1n
        bD.x = s1n;  bD.y = CDEN;                // K0=S1n, K1=F^2*eps+eps
    } else {
        aN.x = 1.0f; aN.y = 0.0f;                // K2=1,   K3=0
        aD.x = 0.0f; aD.y = 0.0f;                // K2=0,   K3=0
        bN.x = CNUM; bN.y = 0.0f;                // K2=F^2*eps^2+eps, K3=0
        bD.x = 0.0f; bD.y = 0.0f;                // K2=0,   K3=0
    }

    v8f cn = {};
    v8f cd = {};
    // num[m,n] = S2m*S2n + 2eps*S1m*S1n + (F^2 eps^2 + eps)
    cn = __builtin_amdgcn_wmma_f32_16x16x4_f32(false, aN, false, bN, (short)0, cn, false, false);
    // den[m,n] = S1m*S1n + (F^2 eps + eps)
    cd = __builtin_amdgcn_wmma_f32_16x16x4_f32(false, aD, false, bD, (short)0, cd, false, false);

    // C/D layout: VGPR r -> M = r + 8*half, N = idx
    const float invld = 1.0f / ldn;
#pragma unroll
    for (int r = 0; r < 8; ++r) {
        const int row = m0 + r + 8 * half;
        const float ln = cn[r] / cd[r] + EPS;    // lehmer_numerators
        out[(size_t)b * CDIM * CDIM + (size_t)row * CDIM + (n0 + idx)] = ln * invld;
    }
}

extern "C" void kernel_launch(void* const* d_in, const int* in_sizes, int n_in,
                              void* d_out, int out_size, void* d_ws, size_t ws_size,
                              hipStream_t stream) {
    (void)in_sizes; (void)n_in; (void)out_size; (void)ws_size;
    const float* x  = (const float*)d_in[0];
    float* out      = (float*)d_out;
    float* ws       = (float*)d_ws;

    float* maxf = ws;                        // BDIM floats (padded to 16)
    float* S1   = ws + 16;                   // BDIM*CDIM
    float* S2   = ws + 16 + BDIM * CDIM;     // BDIM*CDIM
    float* Ld   = ws + 16 + 2 * BDIM * CDIM; // BDIM*CDIM

    k_batch_max<<<BDIM, 256, 0, stream>>>(x, maxf);
    k_channel_sums<<<BDIM * CDIM, 64, 0, stream>>>(x, maxf, S1, S2, Ld);
    dim3 grid(CDIM / 16, CDIM / 16, BDIM);
    k_causality_wmma<<<grid, 32, 0, stream>>>(S1, S2, Ld, out);
}